// Pooling_17093969838316
// MI455X (gfx1250) — compile-verified
//
#include <hip/hip_runtime.h>

typedef float v2f __attribute__((ext_vector_type(2)));
typedef float v8f __attribute__((ext_vector_type(8)));

#define NEG_SLOPE 0.2f
#define POOL_BLOCK 128

// ---------------------------------------------------------------------------
// Kernel 1: score[i] = leakyrelu(dot(feat[i,:], w)) via V_WMMA_F32_16X16X4_F32.
// One wave handles 16 rows; 32 chained K=4 WMMAs cover D=128 in full fp32.
// A layout (ISA 7.12.2, 32-bit A 16x4): lanes 0-15 -> {K0,K1}, lanes 16-31 ->
// {K2,K3}; each lane does one aligned float2 load per K-step. B holds w[k]
// replicated across all 16 columns (same K-to-lane mapping), so every column
// of D equals the score vector; we extract column 0 from lanes 0 and 16.
// Tail handling: rows are CLAMPED to N-1 (loads always in-bounds, EXEC stays
// all-1s through the WMMA chain, no exec-mask juggling in the hot loop);
// out-of-range lanes compute a redundant-but-valid score that is never stored.
// ---------------------------------------------------------------------------
__global__ void __launch_bounds__(256)
score_wmma_kernel(const float* __restrict__ feat, const float* __restrict__ w,
                  float* __restrict__ score, int N, int D) {
    const int lane = threadIdx.x & 31;
    const int wave = threadIdx.x >> 5;
    const long long tileBase =
        ((long long)blockIdx.x * (blockDim.x >> 5) + wave) * 16;

    const int rowInTile = lane & 15;
    const int kAdd = (lane < 16) ? 0 : 2;
    long long row = tileBase + rowInTile;
    if (row > (long long)(N - 1)) row = (long long)(N - 1);   // clamp, not mask
    const float* rowPtr = feat + row * (long long)D;

    v8f c = {};
#pragma unroll 4
    for (int k0 = 0; k0 < D; k0 += 4) {
        const float2 ta = *(const float2*)(rowPtr + k0 + kAdd);
        const float2 tw = *(const float2*)(w + k0 + kAdd);
        v2f a; a[0] = ta.x; a[1] = ta.y;
        v2f b; b[0] = tw.x; b[1] = tw.y;
        // 8 args: (neg_a, A, neg_b, B, c_mod, C, reuse_a, reuse_b)
        c = __builtin_amdgcn_wmma_f32_16x16x4_f32(
            false, a, false, b, (short)0, c, false, false);
    }

    // C/D layout: lane 0 VGPR r = (M=r, N=0); lane 16 VGPR r = (M=8+r, N=0).
    if (lane == 0 || lane == 16) {
        const long long r0 = tileBase + ((lane == 0) ? 0 : 8);
#pragma unroll
        for (int r = 0; r < 8; ++r) {
            const long long rr = r0 + r;
            if (rr < (long long)N) {
                const float v = c[r];
                score[rr] = (v >= 0.f) ? v : NEG_SLOPE * v;
            }
        }
    }
}

// ---------------------------------------------------------------------------
// lower_bound over sorted segment ids
// ---------------------------------------------------------------------------
__device__ __forceinline__ int lower_bound(const int* __restrict__ seg, int n, int key) {
    int lo = 0, hi = n;
    while (lo < hi) {
        const int mid = (lo + hi) >> 1;
        if (seg[mid] < key) lo = mid + 1; else hi = mid;
    }
    return lo;
}

// ---------------------------------------------------------------------------
// Kernel 2: one 128-thread workgroup per segment.
//   phase 1: block max of scores       (LDS tree reduce)
//   phase 2: block sum of exp(s - m)   (LDS tree reduce)
//   phase 3: thread d accumulates sum_i alpha_i * feat[i][d] -> out[b, col+d]
// Deterministic: fixed reduction order, no atomics. Empty segments write 0.
// ---------------------------------------------------------------------------
__global__ void __launch_bounds__(POOL_BLOCK)
pool_kernel(const float* __restrict__ feat, const float* __restrict__ score,
            const int* __restrict__ seg, float* __restrict__ out,
            int colOff, int outStride, int N, int D) {
    const int b = blockIdx.x;
    const int tid = threadIdx.x;
    __shared__ float red[POOL_BLOCK];

    const int start = lower_bound(seg, N, b);
    const int end   = lower_bound(seg, N, b + 1);

    // --- segment max ---
    float m = -3.402823466e38f;
    for (int i = start + tid; i < end; i += POOL_BLOCK)
        m = fmaxf(m, score[i]);
    red[tid] = m;
    __syncthreads();
#pragma unroll
    for (int off = POOL_BLOCK / 2; off > 0; off >>= 1) {
        if (tid < off) red[tid] = fmaxf(red[tid], red[tid + off]);
        __syncthreads();
    }
    m = red[0];
    __syncthreads();

    // --- segment sum of exp ---
    float e = 0.f;
    for (int i = start + tid; i < end; i += POOL_BLOCK)
        e += __expf(score[i] - m);
    red[tid] = e;
    __syncthreads();
#pragma unroll
    for (int off = POOL_BLOCK / 2; off > 0; off >>= 1) {
        if (tid < off) red[tid] += red[tid + off];
        __syncthreads();
    }
    const float s = red[0];
    const float inv_s = (end > start) ? (1.f / s) : 0.f;

    // --- weighted accumulation, coalesced across D ---
    for (int d = tid; d < D; d += POOL_BLOCK) {
        float acc = 0.f;
        for (int i = start; i < end; ++i) {
            // prefetch a few rows ahead (lowers to global_prefetch_b8)
            __builtin_prefetch(feat + (long long)(i + 4) * D + d, 0, 1);
            const float alpha = __expf(score[i] - m) * inv_s;
            acc += alpha * feat[(long long)i * D + d];
        }
        out[(long long)b * outStride + colOff + d] = acc;
    }
}

// ---------------------------------------------------------------------------
// Kernel 3: copy global_feats into out[:, 2D:3D]
// ---------------------------------------------------------------------------
__global__ void __launch_bounds__(256)
concat_global_kernel(const float* __restrict__ g, float* __restrict__ out,
                     long long total, int D, int outStride) {
    const long long idx = (long long)blockIdx.x * blockDim.x + threadIdx.x;
    if (idx < total) {
        const long long b = idx / D;
        const int d = (int)(idx - b * D);
        out[b * outStride + 2 * D + d] = g[idx];
    }
}

extern "C" void kernel_launch(void* const* d_in, const int* in_sizes, int n_in,
                              void* d_out, int out_size, void* d_ws, size_t ws_size,
                              hipStream_t stream) {
    const float* atom_feats   = (const float*)d_in[0];
    const float* bond_feats   = (const float*)d_in[1];
    const float* global_feats = (const float*)d_in[2];
    const float* w_atom       = (const float*)d_in[3];
    const float* w_bond       = (const float*)d_in[4];
    const int*   atom_seg     = (const int*)d_in[5];
    const int*   bond_seg     = (const int*)d_in[6];

    const int D = in_sizes[3];              // w_atom has D elements
    const int N = in_sizes[0] / D;          // 1,000,000
    const int B = in_sizes[2] / D;          // 16,384
    const int outStride = 3 * D;

    float* score_atom = (float*)d_ws;       // N floats
    float* score_bond = score_atom + N;     // N floats  (8 MB total)

    // Scores: one wave per 16 rows, 8 waves per 256-thread block -> 128 rows/block
    const int rowsPerBlock = 128;
    const int gScore = (N + rowsPerBlock - 1) / rowsPerBlock;
    score_wmma_kernel<<<gScore, 256, 0, stream>>>(atom_feats, w_atom, score_atom, N, D);
    score_wmma_kernel<<<gScore, 256, 0, stream>>>(bond_feats, w_bond, score_bond, N, D);

    // Segment softmax + weighted sum, one workgroup per segment
    pool_kernel<<<B, POOL_BLOCK, 0, stream>>>(atom_feats, score_atom, atom_seg,
                                              (float*)d_out, 0, outStride, N, D);
    pool_kernel<<<B, POOL_BLOCK, 0, stream>>>(bond_feats, score_bond, bond_seg,
                                              (float*)d_out, D, outStride, N, D);

    // Concatenate global features
    const long long total = (long long)B * D;
    const int gCopy = (int)((total + 255) / 256);
    concat_global_kernel<<<gCopy, 256, 0, stream>>>(global_feats, (float*)d_out,
                                                    total, D, outStride);
}